// GCNNet_1881195675684
// MI455X (gfx1250) — compile-verified
//
#include <hip/hip_runtime.h>

// ---------------------------------------------------------------------------
// CDNA5 (gfx1250, wave32) GCN pipeline.
// Dense transforms: V_WMMA_F32_16X16X4_F32 with LDS-staged tiles so the inner
// loop is pure ds_load_b64 + v_wmma (no per-element predication).
// ---------------------------------------------------------------------------

typedef __attribute__((ext_vector_type(2))) float v2f;
typedef __attribute__((ext_vector_type(8))) float v8f;

#define K_BLK 32   // K-chunk staged in LDS per iteration (8 WMMA k-steps)

// Workgroup = 4 waves (32,4). Each block computes a 64x64 tile of
// C = A[MxK] * B[KxN]; wave w owns rows [64*bx + 16*w, +16).
// Optional fused epilogue: +bias[n], ReLU, +addend[m,n].
__global__ void __launch_bounds__(128)
gemm_wmma_f32(const float* __restrict__ A, const float* __restrict__ B,
              float* __restrict__ C, int M, int K, int N,
              const float* __restrict__ bias,
              const float* __restrict__ addend, int do_relu) {
    // A tile: 64 rows x K_BLK, rows padded to 34 floats -> conflict-free b64
    // fragment reads (stride 34 gives 16 distinct even bank-pairs mod 64).
    __shared__ float As[64][K_BLK + 2];
    // B tile: K-row pairs interleaved so fragment rows (kk, kk+1) are one b64.
    __shared__ float Bs[K_BLK / 2][64][2];

    const int lane   = threadIdx.x;             // 0..31 (wave32)
    const int wave   = threadIdx.y;             // 0..3
    const int tid    = wave * 32 + lane;        // 0..127
    const int m_blk  = blockIdx.x * 64;         // block row base
    const int m_base = m_blk + wave * 16;       // this wave's row base
    const int n_base = blockIdx.y * 64;

    v8f acc[4] = {{}, {}, {}, {}};

    for (int k0 = 0; k0 < K; k0 += K_BLK) {
        // ---- prefetch next K-chunk while we still have this one to chew on
        if (k0 + K_BLK < K) {
            int pm = m_blk + (tid >> 1);            // 64 rows over 128 threads
            if (pm >= M) pm = M - 1;
            int pk = k0 + K_BLK + (tid & 1) * 16;
            if (pk >= K) pk = K - 1;
            __builtin_prefetch(&A[(size_t)pm * K + pk], 0, 3);
            int bk = k0 + K_BLK + (tid >> 6);
            if (bk >= K) bk = K - 1;
            int bn = n_base + (tid & 63);
            if (bn >= N) bn = N > 0 ? N - 1 : 0;
            __builtin_prefetch(&B[(size_t)bk * N + bn], 0, 3);
        }

        // ---- stage A: 64x32 = 2048 elements, 16 per thread, zero-padded
#pragma unroll
        for (int i = 0; i < 16; ++i) {
            int idx = tid + i * 128;
            int r = idx >> 5, c = idx & 31;
            int gm = m_blk + r, gk = k0 + c;
            As[r][c] = (gm < M && gk < K) ? A[(size_t)gm * K + gk] : 0.0f;
        }
        // ---- stage B: 32x64 = 2048 elements, row-pair interleaved
#pragma unroll
        for (int i = 0; i < 16; ++i) {
            int idx = tid + i * 128;
            int kk = idx >> 6, c = idx & 63;
            int gk = k0 + kk, gn = n_base + c;
            Bs[kk >> 1][c][kk & 1] = (gk < K && gn < N) ? B[(size_t)gk * N + gn]
                                                        : 0.0f;
        }
        __syncthreads();

        // ---- branch-free WMMA burst over the staged chunk
        const int row = lane & 15;
        const int koff = (lane >> 4) << 1;       // +0 (lanes 0-15) / +2 (16-31)
#pragma unroll
        for (int k = 0; k < K_BLK; k += 4) {
            int kk = k + koff;                   // even -> 8B aligned
            v2f a = *(const v2f*)&As[wave * 16 + row][kk];
#pragma unroll
            for (int t = 0; t < 4; ++t) {
                v2f b = *(const v2f*)&Bs[kk >> 1][t * 16 + row][0];
                acc[t] = __builtin_amdgcn_wmma_f32_16x16x4_f32(
                    false, a, false, b, (short)0, acc[t], false, false);
            }
        }
        __syncthreads();
    }

    // ---- epilogue: C/D layout VGPR j -> M = j + (lane<16?0:8), N = lane&15
    const int col     = lane & 15;
    const int row_off = (lane >> 4) * 8;
#pragma unroll
    for (int t = 0; t < 4; ++t) {
        int nn = n_base + t * 16 + col;
        if (nn >= N) continue;
#pragma unroll
        for (int j = 0; j < 8; ++j) {
            int mm = m_base + row_off + j;
            if (mm < M) {
                float v = acc[t][j];
                if (bias)    v += bias[nn];
                if (do_relu) v  = fmaxf(v, 0.0f);
                if (addend)  v += addend[(size_t)mm * N + nn];
                C[(size_t)mm * N + nn] = v;
            }
        }
    }
}

// ------------------------- graph aggregation kernels -----------------------

__global__ void k_deg_init(float* __restrict__ deg, int n) {
    int i = blockIdx.x * blockDim.x + threadIdx.x;
    if (i < n) deg[i] = 1.0f;                    // self-loop contribution
}

__global__ void k_deg_count(const long long* __restrict__ dst, float* __restrict__ deg, int E) {
    int e = blockIdx.x * blockDim.x + threadIdx.x;
    if (e < E) atomicAdd(&deg[(int)dst[e]], 1.0f);
}

__global__ void k_deg_rsqrt(float* __restrict__ deg, int n) {
    int i = blockIdx.x * blockDim.x + threadIdx.x;
    if (i < n) deg[i] = rsqrtf(deg[i]);          // deg >= 1 always (self-loops)
}

// out[i,:] = xw[i,:] * dis[i]^2   (self-loop term; also zero-initializes out)
__global__ void k_agg_init(const float* __restrict__ xw, const float* __restrict__ dis,
                           float* __restrict__ out, int n, int F) {
    size_t idx = (size_t)blockIdx.x * blockDim.x + threadIdx.x;
    if (idx < (size_t)n * F) {
        int i = (int)(idx / F);
        float d = dis[i];
        out[idx] = xw[idx] * d * d;
    }
}

// out[dst,:] += xw[src,:] * dis[src]*dis[dst]   (block per edge, lanes over F)
__global__ void k_agg_edges(const float* __restrict__ xw, const long long* __restrict__ src,
                            const long long* __restrict__ dst, const float* __restrict__ dis,
                            float* __restrict__ out, int F) {
    int e = blockIdx.x;
    int f = blockIdx.y * blockDim.x + threadIdx.x;
    if (f >= F) return;
    int s = (int)src[e];
    int d = (int)dst[e];
    float norm = dis[s] * dis[d];
    atomicAdd(&out[(size_t)d * F + f], xw[(size_t)s * F + f] * norm);
}

__global__ void k_bias_relu(float* __restrict__ h, const float* __restrict__ b, int n, int F) {
    size_t idx = (size_t)blockIdx.x * blockDim.x + threadIdx.x;
    if (idx < (size_t)n * F) {
        int f = (int)(idx % F);
        h[idx] = fmaxf(h[idx] + b[f], 0.0f);
    }
}

__global__ void k_zero_u32(unsigned* __restrict__ p, size_t n) {
    size_t idx = (size_t)blockIdx.x * blockDim.x + threadIdx.x;
    if (idx < n) p[idx] = 0u;
}

// Max-pool per graph.  h >= 0 after ReLU, so IEEE bit patterns order as u32;
// zero-init also reproduces the reference's empty-graph -> 0 guard.
__global__ void k_gmax(const float* __restrict__ h, const long long* __restrict__ batch,
                       unsigned* __restrict__ g, int F) {
    int i = blockIdx.x;
    int f = blockIdx.y * blockDim.x + threadIdx.x;
    if (f >= F) return;
    int gr = (int)batch[i];
    atomicMax(&g[(size_t)gr * F + f], __float_as_uint(h[(size_t)i * F + f]));
}

// ---------------------------------------------------------------------------

extern "C" void kernel_launch(void* const* d_in, const int* in_sizes, int n_in,
                              void* d_out, int out_size, void* d_ws, size_t ws_size,
                              hipStream_t stream) {
    (void)n_in; (void)out_size; (void)ws_size;

    const int F1 = 78, F2 = 156, F3 = 312, FO = 489;

    const float*     x     = (const float*)d_in[0];
    const long long* ei    = (const long long*)d_in[1];   // int64 [2, E]
    const long long* batch = (const long long*)d_in[2];   // int64 [n], sorted
    const float*     drug2 = (const float*)d_in[3];
    const float* W1 = (const float*)d_in[4];   const float* b1  = (const float*)d_in[5];
    const float* W2 = (const float*)d_in[6];   const float* b2  = (const float*)d_in[7];
    const float* W3 = (const float*)d_in[8];   const float* b3  = (const float*)d_in[9];
    const float* Wfc = (const float*)d_in[10]; const float* bfc = (const float*)d_in[11];

    const int n = in_sizes[0] / F1;      // 50000 nodes
    const int E = in_sizes[1] / 2;       // 800000 edges
    const int G = in_sizes[3] / FO;      // 2048 graphs
    const long long* src = ei;
    const long long* dst = ei + E;

    // ---- workspace carve-out (all buffers initialized by our kernels) ----
    auto align256 = [](size_t s) { return (s + 255) & ~(size_t)255; };
    char* p = (char*)d_ws;
    float*    dis = (float*)p;    p += align256((size_t)n * sizeof(float));
    float*    xw  = (float*)p;    p += align256((size_t)n * F3 * sizeof(float));
    float*    hb  = (float*)p;    p += align256((size_t)n * F3 * sizeof(float));
    unsigned* g   = (unsigned*)p; p += align256((size_t)G * F3 * sizeof(float));

    // ---- symmetric-normalization coefficients ----
    k_deg_init <<<(n + 255) / 256, 256, 0, stream>>>(dis, n);
    k_deg_count<<<(E + 255) / 256, 256, 0, stream>>>(dst, dis, E);
    k_deg_rsqrt<<<(n + 255) / 256, 256, 0, stream>>>(dis, n);

    // ---- one GCN layer: GEMM (WMMA) -> self-loop init -> edge scatter -> bias+ReLU
    auto layer = [&](const float* hin, const float* W, const float* b, int Fin, int Fout) {
        dim3 blk(32, 4);
        dim3 grd((n + 63) / 64, (Fout + 63) / 64);
        gemm_wmma_f32<<<grd, blk, 0, stream>>>(hin, W, xw, n, Fin, Fout,
                                               nullptr, nullptr, 0);
        size_t tot = (size_t)n * Fout;
        k_agg_init<<<(unsigned)((tot + 255) / 256), 256, 0, stream>>>(xw, dis, hb, n, Fout);
        dim3 eg((unsigned)E, (Fout + 127) / 128);
        k_agg_edges<<<eg, 128, 0, stream>>>(xw, src, dst, dis, hb, Fout);
        k_bias_relu<<<(unsigned)((tot + 255) / 256), 256, 0, stream>>>(hb, b, n, Fout);
    };

    layer(x,  W1, b1, F1, F1);   // 78  -> 78
    layer(hb, W2, b2, F1, F2);   // 78  -> 156
    layer(hb, W3, b3, F2, F3);   // 156 -> 312

    // ---- global max pool over graphs ----
    size_t gtot = (size_t)G * F3;
    k_zero_u32<<<(unsigned)((gtot + 255) / 256), 256, 0, stream>>>(g, gtot);
    dim3 mg((unsigned)n, (F3 + 127) / 128);
    k_gmax<<<mg, 128, 0, stream>>>(hb, batch, g, F3);

    // ---- FC head with fused epilogue: out = relu(g @ Wfc + bfc) + drug2 ----
    {
        dim3 blk(32, 4);
        dim3 grd((G + 63) / 64, (FO + 63) / 64);
        gemm_wmma_f32<<<grd, blk, 0, stream>>>((const float*)g, Wfc, (float*)d_out,
                                               G, F3, FO, bfc, drug2, 1);
    }
}